// resRGCN_43817256354378
// MI455X (gfx1250) — compile-verified
//
#include <hip/hip_runtime.h>

typedef float v2f __attribute__((ext_vector_type(2)));
typedef float v8f __attribute__((ext_vector_type(8)));

#define NNODES 50000
#define NPER   25000
#define NEDGE  600000
#define NREL   8
#define NBASE  4
#define INDIM  256
#define HIDD   128
#define OUTD   64

// ---------------------------------------------------------------------------
// Generic f32 WMMA GEMM: D = epilogue(A[M,K] @ B[K,Nc])
// MODE 0: D = relu(AB + bias)                     (input projection)
// MODE 1: D = AB                                  (per-relation projection)
// MODE 2: D = relu(acc + AB + bias)               (conv0 epilogue)
// MODE 3: D = res + relu(acc + AB + bias)         (conv1 residual epilogue)
// MODE 4: D = acc + AB + bias                     (final conv, no relu)
// One wave computes one 16x16 C tile; 8 waves (256 thr) stack along M.
// ---------------------------------------------------------------------------
template<int MODE>
__global__ __launch_bounds__(256)
void gemm16_wmma(const float* __restrict__ A, const float* __restrict__ B,
                 const float* __restrict__ bias, const float* __restrict__ acc,
                 const float* __restrict__ res, float* __restrict__ D,
                 int M, int K, int Nc)
{
    const int lane = threadIdx.x & 31;
    const int wave = threadIdx.x >> 5;
    const int l16  = lane & 15;
    const int half = lane >> 4;

    const int m0 = (blockIdx.y * 8 + wave) * 16;
    const int n0 = blockIdx.x * 16;
    if (m0 >= M) return;                       // wave-uniform exit (EXEC stays all-1s)

    int rowA = m0 + l16;                       // A-frag row for this lane
    if (rowA >= M) rowA = M - 1;               // clamp; masked rows dropped at store
    const int col = n0 + l16;                  // B/C column for this lane

    v8f c = {0.f, 0.f, 0.f, 0.f, 0.f, 0.f, 0.f, 0.f};
    for (int k0 = 0; k0 < K; k0 += 4) {
        const int ka = k0 + 2 * half;          // lanes 16-31 hold K+2, K+3
        v2f a = *(const v2f*)(A + (size_t)rowA * K + ka);
        v2f b;
        b.x = B[(size_t)(ka    ) * Nc + col];
        b.y = B[(size_t)(ka + 1) * Nc + col];
        c = __builtin_amdgcn_wmma_f32_16x16x4_f32(false, a, false, b,
                                                  (short)0, c, false, false);
    }

    #pragma unroll
    for (int i = 0; i < 8; ++i) {
        const int row = m0 + half * 8 + i;     // C layout: lanes>=16 hold rows M+8..M+15
        if (row < M) {
            const size_t off = (size_t)row * Nc + col;
            float v = c[i];
            if (MODE == 0) { v += bias[col]; v = v > 0.f ? v : 0.f; }
            if (MODE == 2) { v += acc[off] + bias[col]; v = v > 0.f ? v : 0.f; }
            if (MODE == 3) { v += acc[off] + bias[col]; v = v > 0.f ? v : 0.f; v += res[off]; }
            if (MODE == 4) { v += acc[off] + bias[col]; }
            D[off] = v;
        }
    }
}

// ---------------------------------------------------------------------------
// Small helpers
// ---------------------------------------------------------------------------
__global__ void zero_k(float* __restrict__ p, int n) {
    int i = blockIdx.x * blockDim.x + threadIdx.x;
    if (i < n) p[i] = 0.f;
}

__global__ void count_k(const int* __restrict__ dst, const int* __restrict__ et,
                        float* __restrict__ cnt, int E) {
    int stride = gridDim.x * blockDim.x;
    for (int e = blockIdx.x * blockDim.x + threadIdx.x; e < E; e += stride)
        atomicAdd(&cnt[(size_t)dst[e] * NREL + et[e]], 1.0f);
}

__global__ void invnorm_k(float* __restrict__ p, int n) {
    int i = blockIdx.x * blockDim.x + threadIdx.x;
    if (i < n) p[i] = 1.0f / fmaxf(p[i], 1.0f);
}

// W[r,i,o] = sum_b comp[r,b] * bases[b,i,o]
__global__ void combine_k(const float* __restrict__ bases, const float* __restrict__ comp,
                          float* __restrict__ W, int sz /*HID*Fo*/) {
    int i = blockIdx.x * blockDim.x + threadIdx.x;
    if (i >= NREL * sz) return;
    int r = i / sz, j = i - r * sz;
    float s = 0.f;
    #pragma unroll
    for (int b = 0; b < NBASE; ++b) s += comp[r * NBASE + b] * bases[(size_t)b * sz + j];
    W[i] = s;
}

// one wave per edge; lanes cover the feature dim; acc[dst] += norm * proj[src]
__global__ __launch_bounds__(256)
void scatter_k(const float* __restrict__ proj, const int* __restrict__ src,
               const int* __restrict__ dst, const int* __restrict__ et,
               const float* __restrict__ norm, float* __restrict__ acc,
               int E, int rsel, int F)
{
    const int lane  = threadIdx.x & 31;
    const int wid   = blockIdx.x * (blockDim.x >> 5) + (threadIdx.x >> 5);
    const int nwave = gridDim.x * (blockDim.x >> 5);
    for (int e = wid; e < E; e += nwave) {
        if (et[e] != rsel) continue;
        const int s = src[e], d = dst[e];
        const float w = norm[(size_t)d * NREL + rsel];
        const float* ps = proj + (size_t)s * F;
        float*       pd = acc  + (size_t)d * F;
        for (int c = lane; c < F; c += 32)
            atomicAdd(&pd[c], w * ps[c]);
    }
}

// ---------------------------------------------------------------------------
// Host-side plumbing
// ---------------------------------------------------------------------------
static void launch_gemm(int mode, const float* A, const float* B, const float* bias,
                        const float* acc, const float* res, float* D,
                        int M, int K, int Nc, hipStream_t s)
{
    dim3 grid(Nc / 16, ((M + 15) / 16 + 7) / 8);
    dim3 block(256);
    switch (mode) {
    case 0: gemm16_wmma<0><<<grid, block, 0, s>>>(A, B, bias, acc, res, D, M, K, Nc); break;
    case 1: gemm16_wmma<1><<<grid, block, 0, s>>>(A, B, bias, acc, res, D, M, K, Nc); break;
    case 2: gemm16_wmma<2><<<grid, block, 0, s>>>(A, B, bias, acc, res, D, M, K, Nc); break;
    case 3: gemm16_wmma<3><<<grid, block, 0, s>>>(A, B, bias, acc, res, D, M, K, Nc); break;
    case 4: gemm16_wmma<4><<<grid, block, 0, s>>>(A, B, bias, acc, res, D, M, K, Nc); break;
    }
}

static void run_conv(const float* Xin, const float* bases, const float* comp,
                     const float* root, const float* bias, int Fo,
                     int epi_mode, const float* res, float* Dst,
                     const float* norm, float* acc, float* proj, float* Wrel,
                     const int* src, const int* dst, const int* et, hipStream_t s)
{
    const int sz = HIDD * Fo;
    combine_k<<<(NREL * sz + 255) / 256, 256, 0, s>>>(bases, comp, Wrel, sz);
    zero_k<<<((int)((size_t)NNODES * Fo) + 255) / 256, 256, 0, s>>>(acc, NNODES * Fo);
    for (int r = 0; r < NREL; ++r) {
        launch_gemm(1, Xin, Wrel + (size_t)r * sz, nullptr, nullptr, nullptr,
                    proj, NNODES, HIDD, Fo, s);
        scatter_k<<<2048, 256, 0, s>>>(proj, src, dst, et, norm, acc, NEDGE, r, Fo);
    }
    launch_gemm(epi_mode, Xin, root, bias, acc, res, Dst, NNODES, HIDD, Fo, s);
}

extern "C" void kernel_launch(void* const* d_in, const int* in_sizes, int n_in,
                              void* d_out, int out_size, void* d_ws, size_t ws_size,
                              hipStream_t stream)
{
    const float* x_paper  = (const float*)d_in[0];
    const float* x_author = (const float*)d_in[1];
    const float* W_paper  = (const float*)d_in[2];
    const float* b_paper  = (const float*)d_in[3];
    const float* W_author = (const float*)d_in[4];
    const float* b_author = (const float*)d_in[5];
    const float* bases0 = (const float*)d_in[6];
    const float* comp0  = (const float*)d_in[7];
    const float* root0  = (const float*)d_in[8];
    const float* bias0  = (const float*)d_in[9];
    const float* bases1 = (const float*)d_in[10];
    const float* comp1  = (const float*)d_in[11];
    const float* root1  = (const float*)d_in[12];
    const float* bias1  = (const float*)d_in[13];
    const float* bases2 = (const float*)d_in[14];
    const float* comp2  = (const float*)d_in[15];
    const float* root2  = (const float*)d_in[16];
    const float* bias2  = (const float*)d_in[17];
    const int*   eidx   = (const int*)d_in[18];   // [2, E]
    const int*   etype  = (const int*)d_in[19];   // [E]
    const int* src = eidx;
    const int* dst = eidx + NEDGE;

    // Output layout: out[N,64] ++ X0[N,128] ++ X0[N,128] ++ X1[N,128]
    float* out = (float*)d_out;
    float* X0  = out + (size_t)NNODES * OUTD;
    float* X0b = X0  + (size_t)NNODES * HIDD;
    float* X1  = X0b + (size_t)NNODES * HIDD;

    // Workspace layout (floats): norm | acc | proj | Wrel | X2   (~79 MB total)
    float* ws   = (float*)d_ws;
    float* norm = ws;                                         // N*R
    float* acc  = norm + (size_t)NNODES * NREL;               // N*128
    float* proj = acc  + (size_t)NNODES * HIDD;               // N*128
    float* Wrel = proj + (size_t)NNODES * HIDD;               // R*128*128
    float* X2   = Wrel + (size_t)NREL * HIDD * HIDD;          // N*128

    // 1) per-(dst,relation) mean normalization (shared by all three convs)
    zero_k<<<(NNODES * NREL + 255) / 256, 256, 0, stream>>>(norm, NNODES * NREL);
    count_k<<<2048, 256, 0, stream>>>(dst, etype, norm, NEDGE);
    invnorm_k<<<(NNODES * NREL + 255) / 256, 256, 0, stream>>>(norm, NNODES * NREL);

    // 2) input projections + ReLU -> X0 (concat paper|author), duplicate to X0b
    launch_gemm(0, x_paper,  W_paper,  b_paper,  nullptr, nullptr,
                X0,                        NPER, INDIM, HIDD, stream);
    launch_gemm(0, x_author, W_author, b_author, nullptr, nullptr,
                X0 + (size_t)NPER * HIDD,  NPER, INDIM, HIDD, stream);
    hipMemcpyAsync(X0b, X0, sizeof(float) * (size_t)NNODES * HIDD,
                   hipMemcpyDeviceToDevice, stream);

    // 3) conv0: X1 = relu(rgcn(X0))
    run_conv(X0, bases0, comp0, root0, bias0, HIDD, /*epi=*/2, nullptr, X1,
             norm, acc, proj, Wrel, src, dst, etype, stream);

    // 4) conv1: X2 = X1 + relu(rgcn(X1))
    run_conv(X1, bases1, comp1, root1, bias1, HIDD, /*epi=*/3, X1, X2,
             norm, acc, proj, Wrel, src, dst, etype, stream);

    // 5) conv2: out = rgcn(X2)   (64-wide, no relu)
    run_conv(X2, bases2, comp2, root2, bias2, OUTD, /*epi=*/4, nullptr, out,
             norm, acc, proj, Wrel, src, dst, etype, stream);
}